// FeatureSelector_71373766525208
// MI455X (gfx1250) — compile-verified
//
#include <hip/hip_runtime.h>

// FeatureSelector: gate = clip(mu + 0.1*(noise + 0.25*extra) + 0.5, 0, 1)
// select top-k bands by gate (ascending index order), out = x[:, idx] * gate[idx]
//
// Kernel 1: tiny top-k/compaction (one block).
// Kernel 2: bandwidth-bound gather+scale; the scale is done exactly via
//           v_wmma_f32_16x16x4_f32 as D = (g*I16) x Xtile accumulated over
//           4 K-chunks (each product is g*x + zeros -> bit-exact f32).

typedef float v2f __attribute__((ext_vector_type(2)));
typedef float v8f __attribute__((ext_vector_type(8)));

#define SIGMA 0.1f

__global__ void gate_topk_kernel(const float* __restrict__ mu,
                                 const float* __restrict__ noise,
                                 const float* __restrict__ extra,
                                 int nbands, int k,
                                 int* __restrict__ idx_out,
                                 float* __restrict__ gate_out) {
    __shared__ float sgate[1024];
    __shared__ int   ssel[1024];
    const int t = threadIdx.x;

    for (int b = t; b < nbands; b += blockDim.x) {
        float z = mu[b] + SIGMA * (noise[b] + 0.25f * extra[b]);
        float g = fminf(fmaxf(z + 0.5f, 0.0f), 1.0f);
        sgate[b] = g;
    }
    __syncthreads();

    // rank = #bands strictly greater, ties broken toward lower index
    // (matches jax.lax.top_k ordering); selected iff rank < k.
    for (int b = t; b < nbands; b += blockDim.x) {
        float g = sgate[b];
        int rank = 0;
        for (int j = 0; j < nbands; ++j) {
            float gj = sgate[j];
            rank += (gj > g) || (gj == g && j < b);
        }
        ssel[b] = (rank < k) ? 1 : 0;
    }
    __syncthreads();

    // compact in ascending band order == sorted(idx)
    for (int b = t; b < nbands; b += blockDim.x) {
        if (ssel[b]) {
            int pos = 0;
            for (int j = 0; j < b; ++j) pos += ssel[j];
            idx_out[pos]  = b;
            gate_out[pos] = sgate[b];
        }
    }
}

// One wave (32 lanes) per 16x16 f32 tile; 4 tiles per 32x32 patch.
// D[M][N] = sum_c sum_K A_c[M][K]*B_c[K][N], A_c = g * I16[:, 4c..4c+3].
__global__ void gather_scale_wmma_kernel(const float* __restrict__ x,
                                         const int* __restrict__ idx,
                                         const float* __restrict__ gate,
                                         float* __restrict__ out,
                                         int nbands, int k, int ntiles) {
    const int wavesPerBlock = blockDim.x >> 5;
    const int wave = blockIdx.x * wavesPerBlock + (threadIdx.x >> 5);
    if (wave >= ntiles) return;          // wave-uniform guard, EXEC stays full

    const int lane = threadIdx.x & 31;
    const int col  = lane & 15;          // N for B/D, M for A
    const int h    = lane >> 4;          // lane half selects K offset (+2)

    const int b   = wave / (k * 4);
    const int rem = wave % (k * 4);
    const int j   = rem >> 2;
    const int tt  = rem & 3;
    const int tr  = tt >> 1;
    const int tc  = tt & 1;

    const int   band = idx[j];           // wave-uniform scalar loads
    const float g    = gate[j];

    const float* xb = x   + ((size_t)(b * nbands + band)) * 1024 + (size_t)(tr * 16) * 32 + tc * 16;
    float*       ob = out + ((size_t)(b * k      + j   )) * 1024 + (size_t)(tr * 16) * 32 + tc * 16;

    v8f acc = {0.f, 0.f, 0.f, 0.f, 0.f, 0.f, 0.f, 0.f};

#pragma unroll
    for (int c4 = 0; c4 < 4; ++c4) {
        const int k0 = c4 * 4 + 2 * h;   // this lane-half's K base in chunk c4
        v2f bm, am;
        // B operand: reg r holds X[4c + r + 2h][col]  (64B-contiguous per half-wave)
        bm[0] = xb[(k0 + 0) * 32 + col];
        bm[1] = xb[(k0 + 1) * 32 + col];
        // A operand: g * I16 columns of this chunk; reg r is K = r + 2h, M = col
        am[0] = (col == k0 + 0) ? g : 0.0f;
        am[1] = (col == k0 + 1) ? g : 0.0f;
        acc = __builtin_amdgcn_wmma_f32_16x16x4_f32(
            /*neg_a=*/false, am, /*neg_b=*/false, bm,
            /*c_mod=*/(short)0, acc, /*reuse_a=*/false, /*reuse_b=*/false);
    }

#pragma unroll
    for (int r = 0; r < 8; ++r) {        // D: reg r -> (M = r + 8h, N = col)
        ob[(r + 8 * h) * 32 + col] = acc[r];
    }
}

extern "C" void kernel_launch(void* const* d_in, const int* in_sizes, int n_in,
                              void* d_out, int out_size, void* d_ws, size_t ws_size,
                              hipStream_t stream) {
    const float* x     = (const float*)d_in[0];
    const float* mu    = (const float*)d_in[1];
    const float* noise = (const float*)d_in[2];
    const float* extra = (const float*)d_in[3];
    // d_in[4] is k on device; derive host-side from sizes (graph-capture safe):
    const int       nbands = in_sizes[1];                       // 256
    const long long xsz    = (long long)in_sizes[0];            // B*BANDS*1024
    const int       k      = (int)(((long long)out_size * nbands) / xsz);  // 64
    const int       B      = (int)(xsz / ((long long)nbands * 1024));      // 128

    int*   idx_ws  = (int*)d_ws;
    float* gate_ws = (float*)((char*)d_ws + 1024);

    gate_topk_kernel<<<1, 256, 0, stream>>>(mu, noise, extra, nbands, k,
                                            idx_ws, gate_ws);

    const int ntiles        = B * k * 4;        // 16x16 tiles
    const int threads       = 256;              // 8 waves/block
    const int wavesPerBlock = threads / 32;
    const int blocks        = (ntiles + wavesPerBlock - 1) / wavesPerBlock;
    gather_scale_wmma_kernel<<<blocks, threads, 0, stream>>>(
        x, idx_ws, gate_ws, (float*)d_out, nbands, k, ntiles);
}